// RandomProjectionQuantizer_20263655702835
// MI455X (gfx1250) — compile-verified
//
#include <hip/hip_runtime.h>

typedef float v2f __attribute__((ext_vector_type(2)));
typedef float v8f __attribute__((ext_vector_type(8)));

#define DIMK    1024               // D
#define CDIM    16                 // codebook_dim
#define CSIZE   8192               // codebook_size
#define RPB     64                 // rows per block (4 waves x 16-row tiles)
#define KC      64                 // K chunk staged in LDS
#define NKCH    (DIMK / KC)        // 16
#define CCHUNK  256                // codebook rows staged per chunk
#define NCCH    (CSIZE / CCHUNK)   // 32
#define XPAD    68                 // row pad (mult of 4 dwords, conflict-free)
#define CPAD    20
#define XSZ     (RPB * XPAD)       // 4352 floats per x buffer
#define WSZ     (CDIM * XPAD)      // 1088 floats per W buffer
#define CBSZ    (CCHUNK * CPAD)    // 5120 floats per codebook buffer
#define ZOFF    (2 * XSZ + 2 * WSZ)          // 10880 (phase-3 region reuses [0,ZOFF))
#define SMEMF   (ZOFF + 4 * CDIM * CPAD)     // 12160 floats = 48.6 KB

// CDNA5 async copy: 16B per lane, global -> LDS, tracked by ASYNCcnt.
__device__ __forceinline__ void async_ld16(unsigned lds_byte_addr, const float* gaddr) {
  asm volatile("global_load_async_to_lds_b128 %0, %1, off"
               :: "v"(lds_byte_addr), "v"(gaddr) : "memory");
}
#define WAIT_ASYNC0() asm volatile("s_wait_asynccnt 0x0" ::: "memory")

__global__ __launch_bounds__(128)
void rpq_fused(const float* __restrict__ x, const float* __restrict__ Wm,
               const float* __restrict__ cb, int* __restrict__ out) {
  __shared__ __align__(16) float smem[SMEMF];

  const int t    = threadIdx.x;
  const int lane = t & 31;
  const int wv   = t >> 5;     // wave 0..3
  const int lr   = lane & 15;  // row (A) / col (B,C,D) selector
  const int hi   = lane >> 4;  // half-wave
  const long r0  = (long)blockIdx.x * RPB;

  // ---- async staging helpers (issue only; completion via s_wait_asynccnt) ----
  auto issue_xw = [&](int kc, int buf) {
#pragma unroll
    for (int i = 0; i < 8; ++i) {          // x chunk: 64x64 = 1024 float4
      int f = t + i * 128, row = f >> 4, part = (f & 15) << 2;
      async_ld16((unsigned)(size_t)&smem[buf * XSZ + row * XPAD + part],
                 &x[(r0 + row) * DIMK + kc + part]);
    }
#pragma unroll
    for (int i = 0; i < 2; ++i) {          // W chunk: 16x64 = 256 float4
      int f = t + i * 128, row = f >> 4, part = (f & 15) << 2;
      async_ld16((unsigned)(size_t)&smem[2 * XSZ + buf * WSZ + row * XPAD + part],
                 &Wm[(long)row * DIMK + kc + part]);
    }
  };
  auto issue_cb = [&](int cc0, int buf) {
#pragma unroll
    for (int i = 0; i < 8; ++i) {          // codebook chunk: 256x16 = 1024 float4
      int f = t + i * 128, row = f >> 2, part = (f & 3) << 2;
      async_ld16((unsigned)(size_t)&smem[buf * CBSZ + row * CPAD + part],
                 &cb[(long)(cc0 + row) * CDIM + part]);
    }
  };

  // ---------------- Phase 1: z = x * W^T (f32 WMMA, K=1024, double-buffered) ----------------
  v8f acc = {0.f, 0.f, 0.f, 0.f, 0.f, 0.f, 0.f, 0.f};
  issue_xw(0, 0);
  for (int ki = 0; ki < NKCH; ++ki) {
    WAIT_ASYNC0();                          // own wave's async loads for chunk ki are in LDS
    __syncthreads();                        // everyone's loads visible; prev buffer free
    if (ki + 1 < NKCH) issue_xw((ki + 1) * KC, (ki + 1) & 1);  // overlap with compute below
    const float* xb = &smem[(ki & 1) * XSZ + (wv * 16 + lr) * XPAD];
    const float* wb = &smem[2 * XSZ + (ki & 1) * WSZ + lr * XPAD];
#pragma unroll
    for (int ks = 0; ks < KC / 4; ++ks) {
      const int k0 = ks * 4 + 2 * hi;
      v2f a = *(const v2f*)&xb[k0];         // A[m=lr][k]
      v2f b = *(const v2f*)&wb[k0];         // B[k][n=lr] = W[n][k]
      acc = __builtin_amdgcn_wmma_f32_16x16x4_f32(false, a, false, b,
                                                  (short)0, acc, false, false);
    }
  }
  __syncthreads();                          // all waves done with x/W region before reuse
  issue_cb(0, 0);                           // prefetch codebook chunk 0 (overlaps LayerNorm)

  // ------------- Phase 2: LayerNorm over CD=16 (cols live across 16 lanes) -------------
  float* zsw = &smem[ZOFF + wv * (CDIM * CPAD)];
#pragma unroll
  for (int v = 0; v < 8; ++v) {
    float s = acc[v];
#pragma unroll
    for (int m = 8; m >= 1; m >>= 1) s += __shfl_xor(s, m, 16);
    const float mu = s * (1.0f / 16.0f);
    const float d  = acc[v] - mu;
    float q = d * d;
#pragma unroll
    for (int m = 8; m >= 1; m >>= 1) q += __shfl_xor(q, m, 16);
    const float inv = rsqrtf(q * (1.0f / 16.0f) + 1e-5f);
    // store NEGATED z so the distance WMMA yields 0.5*||c||^2 - z.c (f32 WMMA has no A-neg)
    zsw[(v + 8 * hi) * CPAD + lr] = -(d * inv);
  }

  // ---------------- Phase 3: argmin over codes of 0.5||c||^2 - z.c (double-buffered) ----------------
  v8f minv = {3.4e38f, 3.4e38f, 3.4e38f, 3.4e38f, 3.4e38f, 3.4e38f, 3.4e38f, 3.4e38f};
  int mini[8] = {0, 0, 0, 0, 0, 0, 0, 0};
  float* csq = &smem[2 * CBSZ];             // [256], disjoint from both cb buffers

  for (int ci = 0; ci < NCCH; ++ci) {
    WAIT_ASYNC0();
    __syncthreads();
    if (ci + 1 < NCCH) issue_cb((ci + 1) * CCHUNK, (ci + 1) & 1);  // overlap with compute
    const float* cbb = &smem[(ci & 1) * CBSZ];
#pragma unroll
    for (int i = 0; i < 2; ++i) {           // 0.5*||c||^2 per staged code
      int row = t + i * 128;
      float s = 0.f;
#pragma unroll
      for (int j = 0; j < CDIM; ++j) { float c = cbb[row * CPAD + j]; s += c * c; }
      csq[row] = 0.5f * s;
    }
    __syncthreads();
#pragma unroll 4
    for (int nt = 0; nt < CCHUNK / 16; ++nt) {
      const int c0 = nt * 16;
      const float ch = csq[c0 + lr];        // C-layout: column n == lane
      v8f dd = {ch, ch, ch, ch, ch, ch, ch, ch};
#pragma unroll
      for (int ks = 0; ks < 4; ++ks) {
        const int k0 = ks * 4 + 2 * hi;
        v2f a = *(const v2f*)&zsw[lr * CPAD + k0];        // A[m=lr][k] = -z
        v2f b = *(const v2f*)&cbb[(c0 + lr) * CPAD + k0]; // B[k][n=lr] = c[n][k]
        dd = __builtin_amdgcn_wmma_f32_16x16x4_f32(false, a, false, b,
                                                   (short)0, dd, false, false);
      }
      const int code = ci * CCHUNK + c0 + lr;
#pragma unroll
      for (int v = 0; v < 8; ++v) {
        if (dd[v] < minv[v]) { minv[v] = dd[v]; mini[v] = code; }  // strict < keeps first index
      }
    }
  }

  // cross-lane argmin within each 16-lane group (row m = v + 8*hi), tie -> smaller index
#pragma unroll
  for (int v = 0; v < 8; ++v) {
    float val = minv[v]; int idx = mini[v];
#pragma unroll
    for (int m = 1; m < 16; m <<= 1) {
      float ov = __shfl_xor(val, m, 16);
      int   oi = __shfl_xor(idx, m, 16);
      if (ov < val || (ov == val && oi < idx)) { val = ov; idx = oi; }
    }
    if (lr == 0) out[r0 + wv * 16 + v + 8 * hi] = idx;
  }
}

extern "C" void kernel_launch(void* const* d_in, const int* in_sizes, int n_in,
                              void* d_out, int out_size, void* d_ws, size_t ws_size,
                              hipStream_t stream) {
  (void)in_sizes; (void)n_in; (void)d_ws; (void)ws_size; (void)out_size;
  const float* x  = (const float*)d_in[0];   // (8,2048,1024) f32
  const float* W  = (const float*)d_in[1];   // (16,1024) f32
  const float* cb = (const float*)d_in[2];   // (8192,16) f32
  int* out = (int*)d_out;                    // (8,2048) int32 codes

  const int rows = 8 * 2048;                 // 16384, divisible by RPB
  dim3 grid(rows / RPB), block(128);
  rpq_fused<<<grid, block, 0, stream>>>(x, W, cb, out);
}